// Inferece_PaDiM_1391569404496
// MI455X (gfx1250) — compile-verified
//
#include <hip/hip_runtime.h>
#include <hip/hip_bf16.h>
#include <math.h>

// Problem constants (from reference)
#define Bsz 8
#define Dn  256
#define Pn  3136     // 56*56
#define Hn  56
#define LDA 260      // 256 + 4 floats padding: breaks 64-bank LDS conflicts, keeps 16B align

typedef __attribute__((ext_vector_type(2))) float v2f;
typedef __attribute__((ext_vector_type(8))) float v8f;
typedef __attribute__((ext_vector_type(4))) unsigned int v4u;
typedef __attribute__((ext_vector_type(8))) int v8i;
typedef __attribute__((ext_vector_type(4))) int v4i;

// ---------------------------------------------------------------------------
// Kernel 1: transpose cov (D*D, P) -> covT (P, D*D) so each pixel's 256x256
// matrix is contiguous (256 KB) for the TDM tile load.
// ---------------------------------------------------------------------------
__global__ void cov_transpose_kernel(const float* __restrict__ cov,
                                     float* __restrict__ covT) {
    __shared__ float tile[32][33];
    const int p0 = blockIdx.x * 32;          // pixel tile   (3136 = 98*32)
    const int q0 = blockIdx.y * 32;          // (c*256+d) tile (65536 = 2048*32)
    for (int j = threadIdx.y; j < 32; j += 8) {
        tile[j][threadIdx.x] = cov[(size_t)(q0 + j) * Pn + (p0 + threadIdx.x)];
    }
    __syncthreads();
    for (int j = threadIdx.y; j < 32; j += 8) {
        covT[(size_t)(p0 + j) * 65536 + (q0 + threadIdx.x)] = tile[threadIdx.x][j];
    }
}

// ---------------------------------------------------------------------------
// Kernel 2: one workgroup (256 threads = 8 wave32) per pixel p.
//   1) TDM tensor_load_to_lds: 256x256 f32 tile, HW-padded 4 dwords per row
//      (reproduces LDA=260 in LDS), wave 0 issues, s_wait_tensorcnt 0
//   2) gather delta[b][d] = emb[b, idx[d], p] - mean[d][p]
//   3) blocked right-looking Cholesky, block=16, trailing SYRK update via
//      V_WMMA_F32_16X16X4_F32 (4 chained WMMAs per 16x16 tile)
//   4) per-wave forward substitution L y = delta_b, dist = sqrt(|y|^2)
// ---------------------------------------------------------------------------
__global__ __launch_bounds__(256, 1)
void padim_cholesky_kernel(const float* __restrict__ x1,
                           const float* __restrict__ x2,
                           const float* __restrict__ x3,
                           const int*   __restrict__ idx,
                           const float* __restrict__ mean,
                           const float* __restrict__ cov,
                           const float* __restrict__ covT,
                           int use_t,
                           float* __restrict__ out) {
    __shared__ float As[Dn * LDA];      // 266,240 B
    __shared__ float delta_s[Bsz][Dn];  //   8,192 B
    __shared__ int   idx_s[Dn];         //   1,024 B

    const int tid  = threadIdx.x;
    const int wave = tid >> 5;
    const int lane = tid & 31;
    const int p    = blockIdx.x;
    const int h    = p / Hn, w = p % Hn;

    idx_s[tid] = idx[tid];

    // ---- phase 0: load the 256x256 matrix into LDS -------------------------
    if (use_t) {
        // Tensor Data Mover: one descriptor loads the whole 256KB tile and
        // inserts 4-dword padding after every 256 dwords (row stride -> LDA).
        if (wave == 0) {
            const unsigned long long ga =
                (unsigned long long)(const void*)(covT + (size_t)p * 65536);
            const unsigned int lds_off =
                (unsigned int)(unsigned long long)(const void*)&As[0];

            v4u g0;
            g0[0] = 1u;                                   // count=1 valid D#
            g0[1] = lds_off;                              // lds_addr (bytes)
            g0[2] = (unsigned int)ga;                     // global_addr[31:0]
            g0[3] = (unsigned int)((ga >> 32) & 0x1FFFFFFull)
                    | (2u << 30);                         // addr[56:32] | type=2

            v8i g1;
            g1[0] = (2 << 16)       // data_size = 4B
                  | (1 << 20)       // pad_enable
                  | (7 << 22)       // pad_interval: 256 dwords
                  | (3 << 25);      // pad_amount:   4 dwords (16B)
            g1[1] = 256 << 16;      // tensor_dim0 = 256 (low 16 @ bits 63:48)
            g1[2] = 256 << 16;      // tensor_dim1 = 256 (low 16 @ bits 95:80)
            g1[3] = 256 << 16;      // tile_dim0   = 256 (bits 127:112)
            g1[4] = 256;            // tile_dim1   = 256, tile_dim2 = 0
            g1[5] = 256;            // tensor_dim0_stride = 256 (contiguous)
            g1[6] = 0;              // dim1_stride unused for 2D tile
            g1[7] = 0;

            v4i gz4 = {0, 0, 0, 0};                  // groups 2/3: tensor is 2-D
            v8i gz8 = {0, 0, 0, 0, 0, 0, 0, 0};      // extra group (clang-23 form)
            __builtin_amdgcn_tensor_load_to_lds(g0, g1, gz4, gz4, gz8, 0);
            __builtin_amdgcn_s_wait_tensorcnt(0);
        }
    } else {
        // fallback: direct strided gather from (D,D,P) layout
        for (int l = tid; l < 65536; l += 256) {
            As[(l >> 8) * LDA + (l & 255)] = cov[(size_t)l * Pn + p];
        }
    }
    __syncthreads();

    // ---- phase 1: gather embedding & build delta ---------------------------
    for (int l = tid; l < Bsz * Dn; l += 256) {
        int b = l >> 8, d = l & 255;
        int c = idx_s[d];
        float v;
        if (c < 256) {
            v = x1[(((size_t)b * 256 + c) * Hn + h) * Hn + w];
        } else if (c < 768) {
            v = x2[(((size_t)b * 512 + (c - 256)) * 28 + (h >> 1)) * 28 + (w >> 1)];
        } else {
            v = x3[(((size_t)b * 1024 + (c - 768)) * 14 + (h >> 2)) * 14 + (w >> 2)];
        }
        delta_s[b][d] = v - mean[(size_t)d * Pn + p];
    }

    // ---- phase 2: blocked Cholesky (block = 16, 16 block steps) ------------
    for (int kb = 0; kb < 16; ++kb) {
        const int k0 = kb * 16;

        // 2a) 16x16 diagonal block factor (threads 0..15, thread t = row t)
        for (int j = 0; j < 16; ++j) {
            __syncthreads();
            if (tid == 0) {
                float v = As[(k0 + j) * LDA + k0 + j];
                As[(k0 + j) * LDA + k0 + j] = sqrtf(fmaxf(v, 1e-30f));
            }
            __syncthreads();
            if (tid < 16 && tid > j) {
                float ljj = As[(k0 + j) * LDA + k0 + j];
                As[(k0 + tid) * LDA + k0 + j] /= ljj;
            }
            __syncthreads();
            if (tid < 16 && tid > j) {
                float lij = As[(k0 + tid) * LDA + k0 + j];
                for (int c = j + 1; c <= tid; ++c)
                    As[(k0 + tid) * LDA + k0 + c] -= lij * As[(k0 + c) * LDA + k0 + j];
            }
        }
        __syncthreads();

        // 2b) panel triangular solve: X * Lkk^T = A_panel  (one row per thread)
        {
            int i = k0 + 16 + tid;
            if (i < Dn) {
                #pragma unroll 4
                for (int j = 0; j < 16; ++j) {
                    float s = As[i * LDA + k0 + j];
                    for (int c = 0; c < j; ++c)
                        s -= As[i * LDA + k0 + c] * As[(k0 + j) * LDA + k0 + c];
                    As[i * LDA + k0 + j] = s / As[(k0 + j) * LDA + k0 + j];
                }
            }
        }
        __syncthreads();

        // 2c) trailing update A[ib][jb] -= P_ib * P_jb^T via WMMA f32 16x16x4
        {
            const int nb    = 15 - kb;
            const int ntile = nb * (nb + 1) / 2;
            const int m    = lane & 15;
            const int half = lane >> 4;
            for (int t = wave; t < ntile; t += 8) {           // wave-uniform
                // closed-form inverse of triangular numbering (no divergent loop)
                int mm = (int)((sqrtf(8.0f * (float)t + 1.0f) - 1.0f) * 0.5f);
                if ((mm + 1) * (mm + 2) / 2 <= t) ++mm;       // fp round fix-up
                if (mm * (mm + 1) / 2 > t) --mm;
                const int ib = kb + 1 + mm;
                const int jb = kb + 1 + (t - mm * (mm + 1) / 2);

                const int arow = (ib * 16 + m) * LDA;
                const int brow = (jb * 16 + m) * LDA;
                v8f acc = {0.f, 0.f, 0.f, 0.f, 0.f, 0.f, 0.f, 0.f};
                #pragma unroll
                for (int s = 0; s < 4; ++s) {
                    const int kk = k0 + s * 4;
                    v2f a, bb;
                    // A 16x4: lanes 0-15 M, VGPR0/1 = K {0,1}; lanes 16-31 K {2,3}
                    a[0] = As[arow + kk + 2 * half + 0];
                    a[1] = As[arow + kk + 2 * half + 1];
                    // B 4x16: B[k][n] = P_jb[n][k]; VGPR v, k = v + 2*half
                    bb[0] = As[brow + kk + 2 * half + 0];
                    bb[1] = As[brow + kk + 2 * half + 1];
                    acc = __builtin_amdgcn_wmma_f32_16x16x4_f32(
                        false, a, false, bb, (short)0, acc, false, false);
                }
                // C/D 16x16 f32: VGPR v holds M = v + 8*half, N = lane&15
                #pragma unroll
                for (int v = 0; v < 8; ++v) {
                    As[(ib * 16 + v + 8 * half) * LDA + jb * 16 + m] -= acc[v];
                }
            }
        }
        __syncthreads();
    }

    // ---- phase 3: forward substitution, one wave per RHS b -----------------
    {
        const int b = wave;
        float yloc[8];
        #pragma unroll
        for (int q = 0; q < 8; ++q) yloc[q] = 0.f;
        float dist2 = 0.f;
        for (int i = 0; i < Dn; ++i) {
            float partial = 0.f;
            #pragma unroll
            for (int q = 0; q < 8; ++q) {
                int j = lane + (q << 5);
                float lv = (j < i) ? As[i * LDA + j] : 0.f;
                partial += lv * yloc[q];
            }
            #pragma unroll
            for (int off = 16; off > 0; off >>= 1)
                partial += __shfl_xor(partial, off, 32);
            float y = (delta_s[b][i] - partial) / As[i * LDA + i];
            if (lane == (i & 31)) yloc[i >> 5] = y;
            dist2 += y * y;                    // identical on all lanes
        }
        if (lane == 0)
            out[(size_t)b * Pn + p] = sqrtf(fmaxf(dist2, 0.f));
    }
}

// ---------------------------------------------------------------------------
extern "C" void kernel_launch(void* const* d_in, const int* in_sizes, int n_in,
                              void* d_out, int out_size, void* d_ws, size_t ws_size,
                              hipStream_t stream) {
    const float* x1   = (const float*)d_in[0];
    const float* x2   = (const float*)d_in[1];
    const float* x3   = (const float*)d_in[2];
    const int*   idx  = (const int*)  d_in[3];
    const float* mean = (const float*)d_in[4];
    const float* cov  = (const float*)d_in[5];
    float* out = (float*)d_out;

    const size_t need = (size_t)Pn * 65536 * sizeof(float);   // 822 MB
    const int use_t = (ws_size >= need) ? 1 : 0;
    float* covT = (float*)d_ws;

    if (use_t) {
        cov_transpose_kernel<<<dim3(98, 2048), dim3(32, 8), 0, stream>>>(cov, covT);
    }
    padim_cholesky_kernel<<<Pn, 256, 0, stream>>>(
        x1, x2, x3, idx, mean, cov, covT, use_t, out);
}